// AdaptiveCentralLayer_12936441495585
// MI455X (gfx1250) — compile-verified
//
#include <hip/hip_runtime.h>
#include <hip/hip_bf16.h>

typedef __attribute__((ext_vector_type(2))) float v2f;
typedef __attribute__((ext_vector_type(8))) float v8f;

#define HH 256
#define WW 256
#define NSLICE 1024
#define EPSF 1e-8f

// ---------------------------------------------------------------------------
// Kernel 1: per-slice center-of-mass via WMMA f32 16x16x4 reductions.
// One block (256 thr = 8 waves) per (b,c) slice. Each wave owns 16-row strips.
// For each 16x4 A-chunk of the slice:
//   D_s  += A x ones(4x16)  -> row sums replicated across N
//   D_x  += A x xcol(4x16)  -> x-weighted row sums replicated across N
// A layout (ISA 7.12.2, 32-bit A 16x4): lanes 0-15 -> rows M=lane, K=0,1 in
// VGPR0,1; lanes 16-31 -> rows M=lane-16, K=2,3. So lane loads 2 consecutive
// floats at (row r0+m, col c+2*hi). C/D layout: lane holds D[M][N] with
// N = lane&15, M = v + 8*(lane>>4) for VGPR v.
// ---------------------------------------------------------------------------
__global__ __launch_bounds__(256) void AdaptiveCentral_centroid_kernel(
    const float* __restrict__ kern, int* __restrict__ shifts) {
  __shared__ float acc[3];
  const int tid = threadIdx.x;
  if (tid < 3) acc[tid] = 0.0f;
  __syncthreads();

  const int slice = blockIdx.x;
  const float* base = kern + (size_t)slice * (HH * WW);
  const int lane = tid & 31;
  const int wave = tid >> 5;
  const int m    = lane & 15;   // row within strip
  const int hi   = lane >> 4;   // K-half select (0: K=0,1  1: K=2,3)

  v2f bone; bone.x = 1.0f; bone.y = 1.0f;

  float ps = 0.0f, psy = 0.0f, psx = 0.0f;

  for (int strip = wave; strip < 16; strip += 8) {
    const int r0 = strip * 16;
    const float* rowp = base + (size_t)(r0 + m) * WW;
    v8f ds = {};  // plain row sums
    v8f dx = {};  // x-weighted row sums
    for (int c = 0; c < WW; c += 4) {
      const int col = c + 2 * hi;
      v2f a = *(const v2f*)(rowp + col);           // K[r0+m][col], K[r0+m][col+1]
      v2f bx; bx.x = (float)col; bx.y = (float)(col + 1);
      ds = __builtin_amdgcn_wmma_f32_16x16x4_f32(false, a, false, bone,
                                                 (short)0, ds, false, false);
      dx = __builtin_amdgcn_wmma_f32_16x16x4_f32(false, a, false, bx,
                                                 (short)0, dx, false, false);
    }
#pragma unroll
    for (int v = 0; v < 8; ++v) {
      const float rs = ds[v];                      // rowsum of row r0 + v + 8*hi
      ps  += rs;
      psy += rs * (float)(r0 + v + 8 * hi);
      psx += dx[v];
    }
  }

  // Values are replicated across each 16-lane (N) group; only N==0 contributes.
  if ((tid & 15) == 0) {
    atomicAdd(&acc[0], ps);
    atomicAdd(&acc[1], psy);
    atomicAdd(&acc[2], psx);
  }
  __syncthreads();

  if (tid == 0) {
    const float s  = acc[0] + EPSF;
    const float cy = acc[1] / s;
    const float cx = acc[2] / s;
    // jnp.round is round-half-to-even; rintf uses RNE mode -> matches.
    const int sy = (int)rintf((float)HH * 0.5f - cy);
    const int sx = (int)rintf((float)WW * 0.5f - cx);
    shifts[2 * slice + 0] = sy;
    shifts[2 * slice + 1] = sx;
  }
}

// ---------------------------------------------------------------------------
// Kernel 2: circular roll (gather), out[i][j] = in[(i - sy) & 255][(j - sx) & 255].
// Grid (1024 slices, 16 row-groups), 256 threads, 16 elems/thread; fully
// coalesced 4B/lane loads + stores (128 B/line per wave) -> HBM-bound pass.
// ---------------------------------------------------------------------------
__global__ __launch_bounds__(256) void AdaptiveCentral_roll_kernel(
    const float* __restrict__ in, const int* __restrict__ shifts,
    float* __restrict__ out) {
  const int slice = blockIdx.x;
  const int sy = shifts[2 * slice + 0];
  const int sx = shifts[2 * slice + 1];
  const float* src = in  + (size_t)slice * (HH * WW);
  float*       dst = out + (size_t)slice * (HH * WW);
  const int i0 = blockIdx.y * 16;
#pragma unroll 4
  for (int e = threadIdx.x; e < 16 * WW; e += 256) {
    const int i  = i0 + (e >> 8);
    const int j  = e & (WW - 1);
    const int si = (i - sy) & (HH - 1);   // two's-complement & handles negatives
    const int sj = (j - sx) & (WW - 1);
    dst[(size_t)i * WW + j] = src[(size_t)si * WW + sj];
  }
}

extern "C" void kernel_launch(void* const* d_in, const int* in_sizes, int n_in,
                              void* d_out, int out_size, void* d_ws, size_t ws_size,
                              hipStream_t stream) {
  const float* kern = (const float*)d_in[0];
  float* out = (float*)d_out;
  int* shifts = (int*)d_ws;  // NSLICE * 2 ints = 8 KB

  AdaptiveCentral_centroid_kernel<<<dim3(NSLICE), dim3(256), 0, stream>>>(kern, shifts);
  AdaptiveCentral_roll_kernel<<<dim3(NSLICE, 16), dim3(256), 0, stream>>>(kern, shifts, out);
}